// HierarchicalMKM_Node_26620207301252
// MI455X (gfx1250) — compile-verified
//
#include <hip/hip_runtime.h>
#include <math.h>

// ---------------------------------------------------------------------------
// Contextual Kalman step on MI455X (gfx1250, wave32).
// One wave per batch element: f32 WMMA (V_WMMA_F32_16X16X4_F32) for the three
// dense 32x32x32 matmuls, wave-cooperative register Cholesky + triangular
// solves for S^{-1}, LDS (stride-33 padded) for fragment staging.
// P_post is double-buffered in LDS via GLOBAL_LOAD_ASYNC_TO_LDS_B128 (ASYNCcnt),
// hiding the 4KB/element HBM latency behind the ~8K-instruction compute body.
// Memory roofline: ~8.4KB/element * 65536 ~= 550MB -> ~24us @ 23.3 TB/s.
// ---------------------------------------------------------------------------

typedef float v2f __attribute__((ext_vector_type(2)));
typedef float v8f __attribute__((ext_vector_type(8)));
typedef int   v4i __attribute__((ext_vector_type(4)));

#define LDSP 33            // padded LDS row stride (33 coprime with 64 banks)
#define NW   2             // waves per block
#define BATCH_N 65536

#if defined(__HIP_DEVICE_COMPILE__) &&                                   \
    __has_builtin(__builtin_amdgcn_global_load_async_to_lds_b128) &&     \
    __has_builtin(__builtin_amdgcn_s_wait_asynccnt)
#define USE_ASYNC_LDS 1
#else
#define USE_ASYNC_LDS 0
#endif

// ---- WMMA fragment helpers (layouts per CDNA5 ISA 7.12.2) ------------------
// A (16x4 f32): lanes 0-15 hold M=lane K={kk,kk+1}; lanes 16-31 K={kk+2,kk+3}
template<bool TRA, bool NEGA>
__device__ __forceinline__ v2f load_a_frag(const float* M, int ti, int kk,
                                           int l16, int half) {
  int row = 16 * ti + l16;
  int k0  = kk + 2 * half;
  v2f a;
  if (!TRA) {
    const float* p = M + row * LDSP + k0;
    a.x = p[0]; a.y = p[1];
  } else {  // A[m][k] = M[k][m]
    const float* p = M + k0 * LDSP + row;
    a.x = p[0]; a.y = p[LDSP];
  }
  if (NEGA) { a.x = -a.x; a.y = -a.y; }
  return a;
}

// B (4x16 f32): VGPR0 rows {kk, kk+2}, VGPR1 rows {kk+1, kk+3}, N = lane%16
template<bool TRB>
__device__ __forceinline__ v2f load_b_frag(const float* M, int kk, int tj,
                                           int l16, int half) {
  int n  = 16 * tj + l16;
  int k0 = kk + 2 * half;
  v2f b;
  if (!TRB) {
    b.x = M[k0 * LDSP + n];
    b.y = M[(k0 + 1) * LDSP + n];
  } else {  // B[k][n] = M[n][k]
    b.x = M[n * LDSP + k0];
    b.y = M[n * LDSP + k0 + 1];
  }
  return b;
}

// C/D (16x16 f32): VGPR r -> rows r (lanes 0-15) and r+8 (lanes 16-31)
__device__ __forceinline__ v8f load_c_frag(const float* M, int ti, int tj,
                                           int l16, int half) {
  v8f c;
#pragma unroll
  for (int r = 0; r < 8; ++r)
    c[r] = M[(16 * ti + 8 * half + r) * LDSP + 16 * tj + l16];
  return c;
}

__device__ __forceinline__ void store_c_frag(float* M, int ti, int tj,
                                             int l16, int half, v8f c) {
#pragma unroll
  for (int r = 0; r < 8; ++r)
    M[(16 * ti + 8 * half + r) * LDSP + 16 * tj + l16] = c[r];
}

// D = op(A) * op(B) [+ C], all 32x32, LDS resident, via 32 f32 WMMAs.
template<bool TRA, bool TRB, bool NEGA, bool ADDC>
__device__ __forceinline__ void mm32(float* D, const float* A, const float* B,
                                     const float* C, int lane) {
  const int l16 = lane & 15, half = lane >> 4;
#pragma unroll
  for (int ti = 0; ti < 2; ++ti) {
#pragma unroll
    for (int tj = 0; tj < 2; ++tj) {
      v8f acc;
      if constexpr (ADDC) {
        acc = load_c_frag(C, ti, tj, l16, half);
      } else {
#pragma unroll
        for (int r = 0; r < 8; ++r) acc[r] = 0.0f;
      }
#pragma unroll
      for (int kk = 0; kk < 32; kk += 4) {
        v2f a = load_a_frag<TRA, NEGA>(A, ti, kk, l16, half);
        v2f b = load_b_frag<TRB>(B, kk, tj, l16, half);
        acc = __builtin_amdgcn_wmma_f32_16x16x4_f32(
            false, a, false, b, (short)0, acc, false, false);
      }
      store_c_frag(D, ti, tj, l16, half, acc);
    }
  }
}

#if USE_ASYNC_LDS
// Issue async DMA of one 4KB P matrix (row-major) into a padded LDS tile.
// 8 x GLOBAL_LOAD_ASYNC_TO_LDS_B128: each lane moves 16B per op.
__device__ __forceinline__ void async_stage_P(const float* __restrict__ Pg,
                                              float* Pb, int lane) {
#pragma unroll
  for (int t = 0; t < 8; ++t) {
    int idx = t * 32 + lane;            // 256 float4 chunks
    int row = idx >> 3;
    int c4  = (idx & 7) * 4;
    __attribute__((address_space(1))) v4i* gp =
        (__attribute__((address_space(1))) v4i*)(Pg + idx * 4);
    __attribute__((address_space(3))) v4i* lp =
        (__attribute__((address_space(3))) v4i*)(&Pb[row * LDSP + c4]);
    __builtin_amdgcn_global_load_async_to_lds_b128(gp, lp, 0, 0);
  }
}
#endif

// Synchronous fallback staging through VGPRs (b128 coalesced).
__device__ __forceinline__ void sync_stage_P(const float* __restrict__ Pg,
                                             float* Pb, int lane) {
  const float4* P4 = reinterpret_cast<const float4*>(Pg);
#pragma unroll
  for (int t = 0; t < 8; ++t) {
    int idx = t * 32 + lane;
    int row = idx >> 3;
    int c4  = (idx & 7) * 4;
    float4 v = P4[idx];
    float* p = &Pb[row * LDSP + c4];
    p[0] = v.x; p[1] = v.y; p[2] = v.z; p[3] = v.w;
  }
}

// ---------------------------------------------------------------------------
__global__ __launch_bounds__(NW * 32) void kalman_wmma_kernel(
    const float* __restrict__ x_post, const float* __restrict__ P_post,
    const float* __restrict__ ctx,    const float* __restrict__ meas,
    const float* __restrict__ W1,     const float* __restrict__ b1,
    const float* __restrict__ W2,     const float* __restrict__ b2,
    const float* __restrict__ H,      const float* __restrict__ logQ,
    const float* __restrict__ logR,
    float* __restrict__ x_out, float* __restrict__ P_out,
    int elems_per_wave) {
  // ---- block-shared constants ----
  __shared__ float sH[32 * LDSP];
  __shared__ float sW1[64 * 49];  __shared__ float sB1[64];
  __shared__ float sW2[32 * 65];  __shared__ float sB2[32];
  __shared__ float sQ[32];        __shared__ float sR[32];
  // ---- per-wave working tiles (padded stride 33) ----
  __shared__ float sP [NW][2][32 * LDSP];  // double-buffered P_pred / P_upd
  __shared__ float sT1[NW][32 * LDSP];     // T1 = H * P_pred
  __shared__ float sS [NW][32 * LDSP];     // S, then L (Cholesky), then Z
  __shared__ float sIn[NW][48];
  __shared__ float sHid[NW][64];
  __shared__ float sXp[NW][32];
  __shared__ float sYv[NW][32];
  __shared__ float sDinv[NW][32];

  const int tid = threadIdx.x;
  for (int i = tid; i < 32 * 32; i += NW * 32)
    sH[(i >> 5) * LDSP + (i & 31)] = H[i];
  for (int i = tid; i < 64 * 48; i += NW * 32)
    sW1[(i / 48) * 49 + (i % 48)] = W1[i];
  for (int i = tid; i < 32 * 64; i += NW * 32)
    sW2[(i >> 6) * 65 + (i & 63)] = W2[i];
  if (tid < 64) sB1[tid] = b1[tid];
  if (tid < 32) {
    sB2[tid] = b2[tid];
    sQ[tid]  = expf(logQ[tid]);
    sR[tid]  = expf(logR[tid]) + 1e-6f;   // bake the +eps*I into R
  }
  __syncthreads();

  const int wv   = tid >> 5;
  const int lane = tid & 31;
  float* T1b = sT1[wv];
  float* Sb  = sS[wv];
  float* inb = sIn[wv];
  float* hb  = sHid[wv];
  float* xpb = sXp[wv];
  float* yvb = sYv[wv];
  float* dvb = sDinv[wv];

  const int gwave = blockIdx.x * NW + wv;
  const int base  = gwave * elems_per_wave;

#if USE_ASYNC_LDS
  // Prologue: kick off DMA of the first P matrix into buffer 0.
  async_stage_P(P_post + (size_t)base * 1024, sP[wv][0], lane);
#endif

  for (int e = 0; e < elems_per_wave; ++e) {
    const int b = base + e;
    float* Pb = sP[wv][e & 1];

#if USE_ASYNC_LDS
    // Wait for this element's P DMA, then immediately start the next one
    // into the other buffer so the whole compute body hides its latency.
    __builtin_amdgcn_s_wait_asynccnt(0);
    asm volatile("" ::: "memory");      // keep LDS reads below the wait
    if (e + 1 < elems_per_wave)
      async_stage_P(P_post + (size_t)(b + 1) * 1024, sP[wv][(e + 1) & 1], lane);
    if (e + 2 < elems_per_wave)         // L2 prefetch two ahead
      __builtin_prefetch(P_post + (size_t)(b + 2) * 1024 + lane * 32, 0, 1);
#else
    sync_stage_P(P_post + (size_t)b * 1024, Pb, lane);
    if (e + 1 < elems_per_wave)
      __builtin_prefetch(P_post + (size_t)(b + 1) * 1024 + lane * 32, 0, 1);
#endif
    Pb[lane * LDSP + lane] += sQ[lane];   // P_pred = P_post + diag(exp Q)

    // ---- stage small vectors ----
    inb[lane] = x_post[(size_t)b * 32 + lane];
    if (lane < 16) inb[32 + lane] = ctx[(size_t)b * 16 + lane];
    yvb[lane] = meas[(size_t)b * 32 + lane];     // holds measurement for now

    // ---- MLP transition: h = tanh([x;c] W1^T + b1), x_pred = h W2^T + b2 ----
#pragma unroll
    for (int u = 0; u < 2; ++u) {
      int j = lane + 32 * u;
      float t = sB1[j];
#pragma unroll
      for (int i = 0; i < 48; ++i) t = fmaf(inb[i], sW1[j * 49 + i], t);
      hb[j] = tanhf(t);
    }
    {
      float t = sB2[lane];
#pragma unroll
      for (int j = 0; j < 64; ++j) t = fmaf(hb[j], sW2[lane * 65 + j], t);
      xpb[lane] = t;
    }

    // ---- innovation: y = meas - H x_pred (lane = row) ----
    {
      float acc = yvb[lane];
#pragma unroll
      for (int k = 0; k < 32; ++k) acc = fmaf(-sH[lane * LDSP + k], xpb[k], acc);
      yvb[lane] = acc;
    }

    // ---- T1 = H * P_pred ; S = T1 * H^T + diag(R+eps)  (WMMA) ----
    mm32<false, false, false, false>(T1b, sH, Pb, nullptr, lane);
    mm32<false, true,  false, false>(Sb,  T1b, sH, nullptr, lane);
    Sb[lane * LDSP + lane] += sR[lane];

    // ---- Cholesky S = L L^T, lane = row, columns static in registers ----
    float rw[32];
#pragma unroll
    for (int j = 0; j < 32; ++j) rw[j] = Sb[lane * LDSP + j];
    float mydinv = 0.0f;
#pragma unroll
    for (int k = 0; k < 32; ++k) {
      float akk  = __shfl(rw[k], k);
      float d    = sqrtf(akk);
      float dinv = 1.0f / d;
      mydinv     = (k == lane) ? dinv : mydinv;
      float lik  = rw[k] * dinv;         // L[lane][k]
      rw[k]      = lik;
#pragma unroll
      for (int j = k + 1; j < 32; ++j) {
        float ljk = __shfl(lik, j);      // L[j][k]
        rw[j] = fmaf(-lik, ljk, rw[j]);
      }
    }
#pragma unroll
    for (int j = 0; j < 32; ++j) Sb[lane * LDSP + j] = rw[j];  // L over S
    dvb[lane] = mydinv;

    // ---- solve S Z = T1 (lane = RHS column), L via LDS broadcast ----
    float fy[32];
#pragma unroll
    for (int i = 0; i < 32; ++i) {       // forward: L fy = T1[:,lane]
      float t = T1b[i * LDSP + lane];
#pragma unroll
      for (int k = 0; k < i; ++k) t = fmaf(-Sb[i * LDSP + k], fy[k], t);
      fy[i] = t * dvb[i];
    }
    float z[32];
#pragma unroll
    for (int i = 31; i >= 0; --i) {      // backward: L^T z = fy
      float t = fy[i];
#pragma unroll
      for (int k = i + 1; k < 32; ++k) t = fmaf(-Sb[k * LDSP + i], z[k], t);
      z[i] = t * dvb[i];
    }

    // ---- x_upd = x_pred + K y ;  K[d][i] = Z[i][d] = z[i] on lane d ----
    {
      float xu = xpb[lane];
#pragma unroll
      for (int i = 0; i < 32; ++i) xu = fmaf(z[i], yvb[i], xu);
      x_out[(size_t)b * 32 + lane] = xu;
    }

    // ---- P_upd = P_pred - Z^T T1  (WMMA: A=-Z^T, C=P_pred, D->P) ----
#pragma unroll
    for (int i = 0; i < 32; ++i) Sb[i * LDSP + lane] = z[i];   // Z over L
    mm32<true, false, true, true>(Pb, Sb, T1b, Pb, lane);

    // ---- store P_upd (b128 coalesced) ----
    {
      float4* Pg = reinterpret_cast<float4*>(P_out + (size_t)b * 1024);
#pragma unroll
      for (int t = 0; t < 8; ++t) {
        int idx = t * 32 + lane;
        int row = idx >> 3;
        int c4  = (idx & 7) * 4;
        const float* p = &Pb[row * LDSP + c4];
        Pg[idx] = make_float4(p[0], p[1], p[2], p[3]);
      }
    }
  }
}

// ---------------------------------------------------------------------------
extern "C" void kernel_launch(void* const* d_in, const int* in_sizes, int n_in,
                              void* d_out, int out_size, void* d_ws,
                              size_t ws_size, hipStream_t stream) {
  const float* x_post = (const float*)d_in[0];
  const float* P_post = (const float*)d_in[1];
  const float* ctx    = (const float*)d_in[2];
  const float* meas   = (const float*)d_in[3];
  const float* W1     = (const float*)d_in[4];
  const float* b1     = (const float*)d_in[5];
  const float* W2     = (const float*)d_in[6];
  const float* b2     = (const float*)d_in[7];
  const float* H      = (const float*)d_in[8];
  const float* logQ   = (const float*)d_in[9];
  const float* logR   = (const float*)d_in[10];

  float* x_out = (float*)d_out;                       // [65536, 32]
  float* P_out = x_out + (size_t)BATCH_N * 32;        // [65536, 32, 32]

  const int blocks = 1024;                             // 2048 waves total
  const int elems_per_wave = BATCH_N / (blocks * NW);  // = 32, exact
  hipLaunchKernelGGL(kalman_wmma_kernel, dim3(blocks), dim3(NW * 32), 0, stream,
                     x_post, P_post, ctx, meas, W1, b1, W2, b2, H, logQ, logR,
                     x_out, P_out, elems_per_wave);
}